// multiclusterloss_49057116455286
// MI455X (gfx1250) — compile-verified
//
#include <hip/hip_runtime.h>
#include <math.h>

typedef __attribute__((ext_vector_type(2))) float v2f;
typedef __attribute__((ext_vector_type(8))) float v8f;

#define NROWS 4096
#define INDIM 512
#define HID   256
#define KCL   8
#define DSUBN 400
#define CD    3200
#define ALPHAF 0.001f
#define ETAF   2.0f
#define BETAF  1.0f

static __device__ __forceinline__ v8f wmma4(v2f a, v2f b, v8f c) {
  // D = A(16x4,f32) * B(4x16,f32) + C(16x16,f32)
  return __builtin_amdgcn_wmma_f32_16x16x4_f32(false, a, false, b, (short)0, c,
                                               false, false);
}

// Accumulate a (16*MT) x (16*NT) tile of A(MxK,row-major) @ B(KxN,row-major).
// A fragments: contiguous b64 per row-tile; B fragments: 2 strided b32 per col-tile.
template <int MT, int NT>
static __device__ __forceinline__ void gemm_nn_mt(const float* __restrict__ A, int lda,
                                                  const float* __restrict__ B, int ldb,
                                                  int row0, int col0, int K, int lane,
                                                  v8f acc[MT][NT]) {
  const int half = lane >> 4, l = lane & 15;
  const float* ap[MT];
#pragma unroll
  for (int m = 0; m < MT; ++m)
    ap[m] = A + (size_t)(row0 + m * 16 + l) * lda + 2 * half;
  for (int k0 = 0; k0 < K; k0 += 4) {
    v2f a[MT];
#pragma unroll
    for (int m = 0; m < MT; ++m) a[m] = *(const v2f*)(ap[m] + k0);
    const float* bp = B + (size_t)(k0 + 2 * half) * ldb + col0 + l;
#pragma unroll
    for (int t = 0; t < NT; ++t) {
      v2f b;
      b.x = bp[t * 16];                          // B[k0+2h  ][col]
      b.y = bp[t * 16 + ldb];                    // B[k0+2h+1][col]
#pragma unroll
      for (int m = 0; m < MT; ++m)
        acc[m][t] = wmma4(a[m], b, acc[m][t]);
    }
  }
}

// Accumulate a (16*MT) x (16*NT) tile of A(MxK) @ Bt^T, Bt is (NxK,row-major).
// Both fragment types are contiguous b64 loads.
template <int MT, int NT>
static __device__ __forceinline__ void gemm_nt_mt(const float* __restrict__ A, int lda,
                                                  const float* __restrict__ Bt, int ldbt,
                                                  int row0, int col0, int K, int lane,
                                                  v8f acc[MT][NT]) {
  const int half = lane >> 4, l = lane & 15;
  const float* ap[MT];
#pragma unroll
  for (int m = 0; m < MT; ++m)
    ap[m] = A + (size_t)(row0 + m * 16 + l) * lda + 2 * half;
  const float* bp[NT];
#pragma unroll
  for (int t = 0; t < NT; ++t)
    bp[t] = Bt + (size_t)(col0 + t * 16 + l) * ldbt + 2 * half;
  for (int k0 = 0; k0 < K; k0 += 4) {
    v2f a[MT];
#pragma unroll
    for (int m = 0; m < MT; ++m) a[m] = *(const v2f*)(ap[m] + k0);
#pragma unroll
    for (int t = 0; t < NT; ++t) {
      v2f b = *(const v2f*)(bp[t] + k0);
#pragma unroll
      for (int m = 0; m < MT; ++m)
        acc[m][t] = wmma4(a[m], b, acc[m][t]);
    }
  }
}

// ---------------- h = leaky_relu(x @ W1 + b1)  [4096 x 256] -----------------
__global__ void k_h(const float* __restrict__ x, const float* __restrict__ W1,
                    const float* __restrict__ b1, float* __restrict__ h) {
  int w = blockIdx.x * 4 + (threadIdx.x >> 5);   // (4096/32)*(256/64) = 512 waves
  int lane = threadIdx.x & 31;
  const int NTN = HID / 64;                      // 4
  int mt = w / NTN, nt = w % NTN;
  int row0 = mt * 32, col0 = nt * 64;
  v8f acc[2][4] = {};
  gemm_nn_mt<2, 4>(x, INDIM, W1, HID, row0, col0, INDIM, lane, acc);
  int half = lane >> 4, l = lane & 15;
#pragma unroll
  for (int m = 0; m < 2; ++m)
#pragma unroll
    for (int t = 0; t < 4; ++t)
#pragma unroll
      for (int i = 0; i < 8; ++i) {
        int r = row0 + m * 16 + i + 8 * half, c = col0 + t * 16 + l;
        float v = acc[m][t][i] + b1[c];
        h[(size_t)r * HID + c] = v > 0.f ? v : 0.01f * v;
      }
}

// ------- z = leaky_relu(h @ W2 + b2) + x @ Wsc + bsc  [4096 x 3200] ---------
__global__ void k_z(const float* __restrict__ h, const float* __restrict__ W2,
                    const float* __restrict__ b2, const float* __restrict__ x,
                    const float* __restrict__ Wsc, const float* __restrict__ bsc,
                    float* __restrict__ z) {
  int w = blockIdx.x * 4 + (threadIdx.x >> 5);
  int lane = threadIdx.x & 31;
  const int NTN = CD / 64;                       // 50
  int mt = w / NTN, nt = w % NTN;
  int row0 = mt * 16, col0 = nt * 64;
  v8f acc[1][4] = {}, acc2[1][4] = {};
  gemm_nn_mt<1, 4>(h, HID, W2, CD, row0, col0, HID, lane, acc);
  gemm_nn_mt<1, 4>(x, INDIM, Wsc, CD, row0, col0, INDIM, lane, acc2);
  int half = lane >> 4, l = lane & 15;
#pragma unroll
  for (int t = 0; t < 4; ++t)
#pragma unroll
    for (int i = 0; i < 8; ++i) {
      int r = row0 + i + 8 * half, c = col0 + t * 16 + l;
      float v = acc[0][t][i] + b2[c];
      v = v > 0.f ? v : 0.01f * v;
      z[(size_t)r * CD + c] = v + acc2[0][t][i] + bsc[c];
    }
}

// ------------------- zero accumulators (ws is poisoned) ---------------------
__global__ void k_zero(float* __restrict__ colq, float* __restrict__ cols,
                       float* __restrict__ accs) {
  int t = threadIdx.x;
  if (t < 8) { colq[t] = 0.f; cols[t] = 0.f; }
  if (t < 4) accs[t] = 0.f;
}

// ------------------------ Csq[c] = sum_j C[c][j]^2 --------------------------
__global__ void k_csq(const float* __restrict__ C, float* __restrict__ csq) {
  __shared__ float sm[128];
  int c = blockIdx.x, tid = threadIdx.x;
  float a = 0.f;
  for (int j = tid; j < CD; j += 128) { float v = C[c * CD + j]; a += v * v; }
  sm[tid] = a; __syncthreads();
  for (int st = 64; st > 0; st >>= 1) {
    if (tid < st) sm[tid] += sm[tid + st];
    __syncthreads();
  }
  if (tid == 0) csq[c] = sm[0];
}

// ----- q (Student-t soft assignment), row-normalized + column sums ----------
__global__ void k_q(const float* __restrict__ z, const float* __restrict__ C,
                    const float* __restrict__ csq, float* __restrict__ q,
                    float* __restrict__ colq) {
  __shared__ float sm[9 * 128];
  int row = blockIdx.x, tid = threadIdx.x;
  const float* zr = z + (size_t)row * CD;
  float z2 = 0.f, dot[8] = {};
  for (int j = tid; j < CD; j += 128) {
    float v = zr[j];
    z2 += v * v;
#pragma unroll
    for (int c = 0; c < 8; ++c) dot[c] += v * C[c * CD + j];
  }
  sm[tid] = z2;
#pragma unroll
  for (int c = 0; c < 8; ++c) sm[(c + 1) * 128 + tid] = dot[c];
  __syncthreads();
  for (int st = 64; st > 0; st >>= 1) {
    if (tid < st)
#pragma unroll
      for (int jj = 0; jj < 9; ++jj) sm[jj * 128 + tid] += sm[jj * 128 + tid + st];
    __syncthreads();
  }
  if (tid == 0) {
    float t8[8], ssum = 0.f;
#pragma unroll
    for (int c = 0; c < 8; ++c) {
      float dist2 = sm[0] + csq[c] - 2.f * sm[(c + 1) * 128];
      float t = powf(1.f + dist2 / ALPHAF, -0.5f * (ALPHAF + 1.f));
      t8[c] = t; ssum += t;
    }
#pragma unroll
    for (int c = 0; c < 8; ++c) {
      float qv = t8[c] / ssum;
      q[row * 8 + c] = qv;
      atomicAdd(&colq[c], qv);
    }
  }
}

// ---- s_raw[row][cl] = sum over cluster-cl columns of (z@D)^2, fused --------
// One wave owns 32 rows x one 400-column cluster: zero atomics, zD never stored.
__global__ void k_zd(const float* __restrict__ z, const float* __restrict__ D,
                     float* __restrict__ sraw) {
  int w = blockIdx.x * 4 + (threadIdx.x >> 5);   // (4096/32)*8 = 1024 waves
  int lane = threadIdx.x & 31;
  int mt = w >> 3, cl = w & 7;
  int row0 = mt * 32;
  int half = lane >> 4, l = lane & 15;
  float ss[2][8] = {};
  for (int ct = 0; ct < 5; ++ct) {               // 5 macro-tiles of 80 columns
    int col0 = cl * DSUBN + ct * 80;
    v8f acc[2][5] = {};
    gemm_nn_mt<2, 5>(z, CD, D, CD, row0, col0, CD, lane, acc);
#pragma unroll
    for (int m = 0; m < 2; ++m)
#pragma unroll
      for (int t = 0; t < 5; ++t)
#pragma unroll
        for (int i = 0; i < 8; ++i) ss[m][i] += acc[m][t][i] * acc[m][t][i];
  }
#pragma unroll
  for (int m = 0; m < 2; ++m)
#pragma unroll
    for (int i = 0; i < 8; ++i) {                // reduce over 16 columns/half
      float v = ss[m][i];
      v += __shfl_xor(v, 1); v += __shfl_xor(v, 2);
      v += __shfl_xor(v, 4); v += __shfl_xor(v, 8);
      if (l == 0)                                // lanes 0 and 16 write
        sraw[(row0 + m * 16 + i + 8 * half) * 8 + cl] = v;
    }
}

// ------------- s finalize: scale, row-normalize, column sums ----------------
__global__ void k_sfin(const float* __restrict__ sraw, float* __restrict__ s,
                       float* __restrict__ cols) {
  int row = blockIdx.x * blockDim.x + threadIdx.x;
  if (row >= NROWS) return;
  float t8[8], sum = 0.f;
#pragma unroll
  for (int c = 0; c < 8; ++c) {
    float v = (sraw[row * 8 + c] + ETAF * (float)DSUBN) /
              ((ETAF + 1.f) * (float)DSUBN);
    t8[c] = v; sum += v;
  }
#pragma unroll
  for (int c = 0; c < 8; ++c) {
    float sv = t8[c] / sum;
    s[row * 8 + c] = sv;
    atomicAdd(&cols[c], sv);
  }
}

// ---- loss_d2: sum of squares of off-block-diagonal entries of D@D^T --------
// 32x64 macro-tiles; 16x16 subtiles never straddle a 400-wide cluster block,
// so masking is exact at subtile granularity during the square-accumulate.
__global__ void k_d2(const float* __restrict__ D, float* __restrict__ accs) {
  int w = blockIdx.x * 4 + (threadIdx.x >> 5);   // 100*50 = 5000 macro-tiles
  int lane = threadIdx.x & 31;
  int tr = w / (CD / 64), tc = w % (CD / 64);
  int row0 = tr * 32, col0 = tc * 64;
  int rc[2], cc[4];
#pragma unroll
  for (int m = 0; m < 2; ++m) rc[m] = (row0 + m * 16) / DSUBN;
#pragma unroll
  for (int t = 0; t < 4; ++t) cc[t] = (col0 + t * 16) / DSUBN;
  bool alldiag = true;
#pragma unroll
  for (int m = 0; m < 2; ++m)
#pragma unroll
    for (int t = 0; t < 4; ++t)
      if (rc[m] != cc[t]) alldiag = false;
  if (alldiag) return;                           // fully masked macro-tile
  v8f acc[2][4] = {};
  gemm_nt_mt<2, 4>(D, CD, D, CD, row0, col0, CD, lane, acc);
  float ssum = 0.f;
#pragma unroll
  for (int m = 0; m < 2; ++m)
#pragma unroll
    for (int t = 0; t < 4; ++t)
      if (rc[m] != cc[t])                        // off-block subtiles only
#pragma unroll
        for (int i = 0; i < 8; ++i) ssum += acc[m][t][i] * acc[m][t][i];
  ssum += __shfl_xor(ssum, 1);  ssum += __shfl_xor(ssum, 2);
  ssum += __shfl_xor(ssum, 4);  ssum += __shfl_xor(ssum, 8);
  ssum += __shfl_xor(ssum, 16);
  if (lane == 0) atomicAdd(&accs[1], ssum);
}

// -------------- loss_d1: sum over rows of (||D_i||^2 - 1)^2 -----------------
__global__ void k_d1(const float* __restrict__ D, float* __restrict__ accs) {
  __shared__ float sm[128];
  int r = blockIdx.x, tid = threadIdx.x;
  float a = 0.f;
  for (int j = tid; j < CD; j += 128) { float v = D[(size_t)r * CD + j]; a += v * v; }
  sm[tid] = a; __syncthreads();
  for (int st = 64; st > 0; st >>= 1) {
    if (tid < st) sm[tid] += sm[tid + st];
    __syncthreads();
  }
  if (tid == 0) { float d = sm[0] - 1.f; atomicAdd(&accs[0], d * d); }
}

// ---------------------- KL(p_eu || q) + KL(p_sub || s) ----------------------
__global__ void k_kl(const float* __restrict__ q, const float* __restrict__ s,
                     const float* __restrict__ colq, const float* __restrict__ cols,
                     float* __restrict__ accs) {
  int row = blockIdx.x * blockDim.x + threadIdx.x;
  if (row >= NROWS) return;
  float kl = 0.f;
  {
    float wv[8], wsum = 0.f;
#pragma unroll
    for (int c = 0; c < 8; ++c) { float v = q[row * 8 + c]; wv[c] = v * v / colq[c]; wsum += wv[c]; }
#pragma unroll
    for (int c = 0; c < 8; ++c) { float p = wv[c] / wsum; kl += p * (logf(p) - logf(q[row * 8 + c])); }
  }
  {
    float wv[8], wsum = 0.f;
#pragma unroll
    for (int c = 0; c < 8; ++c) { float v = s[row * 8 + c]; wv[c] = v * v / cols[c]; wsum += wv[c]; }
#pragma unroll
    for (int c = 0; c < 8; ++c) { float p = wv[c] / wsum; kl += p * (logf(p) - logf(s[row * 8 + c])); }
  }
  atomicAdd(&accs[2], kl);
}

__global__ void k_final(const float* __restrict__ accs, float* __restrict__ out) {
  if (threadIdx.x == 0)
    out[0] = ALPHAF * (sqrtf(accs[0]) + sqrtf(accs[1])) + BETAF * accs[2];
}

extern "C" void kernel_launch(void* const* d_in, const int* in_sizes, int n_in,
                              void* d_out, int out_size, void* d_ws, size_t ws_size,
                              hipStream_t stream) {
  const float* x   = (const float*)d_in[0];
  const float* W1  = (const float*)d_in[1];
  const float* b1  = (const float*)d_in[2];
  const float* W2  = (const float*)d_in[3];
  const float* b2  = (const float*)d_in[4];
  const float* Wsc = (const float*)d_in[5];
  const float* bsc = (const float*)d_in[6];
  const float* C   = (const float*)d_in[7];
  const float* D   = (const float*)d_in[8];
  float* out = (float*)d_out;

  float* ws   = (float*)d_ws;
  float* h    = ws;                              // 4096*256
  float* z    = h + (size_t)NROWS * HID;         // 4096*3200
  float* q    = z + (size_t)NROWS * CD;          // 4096*8
  float* s    = q + (size_t)NROWS * KCL;         // 4096*8
  float* sraw = s + (size_t)NROWS * KCL;         // 4096*8
  float* csq  = sraw + (size_t)NROWS * KCL;      // 8
  float* colq = csq + 8;                         // 8
  float* cols = colq + 8;                        // 8
  float* accs = cols + 8;                        // 4: d1, d2, kl

  hipLaunchKernelGGL(k_zero, dim3(1), dim3(32), 0, stream, colq, cols, accs);
  hipLaunchKernelGGL(k_csq, dim3(KCL), dim3(128), 0, stream, C, csq);
  hipLaunchKernelGGL(k_h, dim3((NROWS / 32) * (HID / 64) / 4), dim3(128), 0, stream,
                     x, W1, b1, h);
  hipLaunchKernelGGL(k_z, dim3((NROWS / 16) * (CD / 64) / 4), dim3(128), 0, stream,
                     h, W2, b2, x, Wsc, bsc, z);
  hipLaunchKernelGGL(k_q, dim3(NROWS), dim3(128), 0, stream, z, C, csq, q, colq);
  hipLaunchKernelGGL(k_zd, dim3((NROWS / 32) * KCL / 4), dim3(128), 0, stream,
                     z, D, sraw);
  hipLaunchKernelGGL(k_sfin, dim3(NROWS / 128), dim3(128), 0, stream, sraw, s, cols);
  hipLaunchKernelGGL(k_d2, dim3((CD / 32) * (CD / 64) / 4), dim3(128), 0, stream,
                     D, accs);
  hipLaunchKernelGGL(k_d1, dim3(CD), dim3(128), 0, stream, D, accs);
  hipLaunchKernelGGL(k_kl, dim3(NROWS / 128), dim3(128), 0, stream, q, s, colq,
                     cols, accs);
  hipLaunchKernelGGL(k_final, dim3(1), dim3(1), 0, stream, accs, out);
}